// Encoder_80874234183757
// MI455X (gfx1250) — compile-verified
//
#include <hip/hip_runtime.h>
#include <hip/hip_bf16.h>

typedef __attribute__((ext_vector_type(2))) float v2f;
typedef __attribute__((ext_vector_type(8))) float v8f;

#define N_NODES 100000
#define N_EDGES 1600000
#define F_IN    128
#define F_OUT   64

// ---------------------------------------------------------------------------
// Zero-fill (float4 wide)
// ---------------------------------------------------------------------------
__global__ void zero_kernel(float* __restrict__ p, long long n4) {
    long long i = (long long)blockIdx.x * blockDim.x + threadIdx.x;
    if (i < n4) ((float4*)p)[i] = make_float4(0.f, 0.f, 0.f, 0.f);
}

// ---------------------------------------------------------------------------
// Fused dual GEMM:  P = A @ Wl,  R = A @ Wr      (A: [N_NODES, K], W: [K, 64])
// block = 128 threads (4 waves); each block does 16 rows x 64 cols;
// wave w covers columns [16*w, 16*w+16). Uses V_WMMA_F32_16X16X4_F32 (fp32).
// N_NODES % 16 == 0 so no divergence -> EXEC all ones at every WMMA.
// ---------------------------------------------------------------------------
template <int K>
__global__ __launch_bounds__(128) void dual_gemm_kernel(
    const float* __restrict__ A,
    const float* __restrict__ Wl,
    const float* __restrict__ Wr,
    float* __restrict__ P,
    float* __restrict__ R) {
    const int lane = threadIdx.x & 31;
    const int wave = threadIdx.x >> 5;          // 0..3 -> 16-col slice
    const int half = lane >> 4;                 // 0 or 1
    const int m16  = lane & 15;                 // row-in-tile for A, col-in-tile for B/C
    const int row0 = blockIdx.x * 16;
    const int col  = wave * 16 + m16;           // global output column (0..63)

    const float* __restrict__ arow = A + (long long)(row0 + m16) * K;

    v8f accP = {};
    v8f accR = {};

#pragma unroll 4
    for (int k0 = 0; k0 < K; k0 += 4) {
        const int kk = k0 + half * 2;           // even -> 8B-aligned float2 load
        const float2 av = *(const float2*)(arow + kk);
        v2f a;  a[0] = av.x;           a[1] = av.y;
        v2f bl; bl[0] = Wl[kk * F_OUT + col]; bl[1] = Wl[(kk + 1) * F_OUT + col];
        v2f br; br[0] = Wr[kk * F_OUT + col]; br[1] = Wr[(kk + 1) * F_OUT + col];
        accP = __builtin_amdgcn_wmma_f32_16x16x4_f32(false, a, false, bl,
                                                     (short)0, accP, false, false);
        accR = __builtin_amdgcn_wmma_f32_16x16x4_f32(false, a, false, br,
                                                     (short)0, accR, false, false);
    }

#pragma unroll
    for (int v = 0; v < 8; ++v) {
        const int mr = v + half * 8;            // row within the 16-row tile
        const long long o = (long long)(row0 + mr) * F_OUT + col;
        P[o] = accP[v];
        R[o] = accR[v];
    }
}

// ---------------------------------------------------------------------------
// Edge scatter-add: sums[dst] += feat[src]  (64 floats / edge, float4 loads,
// fp32 global atomics -> L2-resident). Optionally counts in-degree.
// 16 threads per edge (one float4 each), consecutive lanes -> coalesced 16B.
// ---------------------------------------------------------------------------
template <bool COUNT>
__global__ __launch_bounds__(256) void scatter_kernel(
    const float* __restrict__ feat,
    const int* __restrict__ src,
    const int* __restrict__ dst,
    float* __restrict__ sums,
    float* __restrict__ cnt) {
    const long long tid = (long long)blockIdx.x * blockDim.x + threadIdx.x;
    const long long e = tid >> 4;               // edge id
    const int q = (int)(tid & 15);              // float4 slot within 64 floats
    if (e >= N_EDGES) return;
    const int s = src[e];
    const int d = dst[e];
    const float4 v = ((const float4*)(feat + (long long)s * F_OUT))[q];
    float* o = sums + (long long)d * F_OUT + q * 4;
    atomicAdd(o + 0, v.x);
    atomicAdd(o + 1, v.y);
    atomicAdd(o + 2, v.z);
    atomicAdd(o + 3, v.w);
    if (COUNT && q == 0) atomicAdd(cnt + d, 1.0f);
}

// ---------------------------------------------------------------------------
// Combine: out = [relu]( sums/max(cnt,1) + r + bias )
// ---------------------------------------------------------------------------
template <bool RELU>
__global__ __launch_bounds__(256) void combine_kernel(
    const float* __restrict__ sums,
    const float* __restrict__ cnt,
    const float* __restrict__ r,
    const float* __restrict__ bias,
    float* __restrict__ out) {
    const long long i = (long long)blockIdx.x * blockDim.x + threadIdx.x;
    if (i >= (long long)N_NODES * F_OUT) return;
    const int row = (int)(i >> 6);
    const int c   = (int)(i & 63);
    const float mean = sums[i] / fmaxf(cnt[row], 1.0f);
    float v = mean + r[i] + bias[c];
    if (RELU) v = fmaxf(v, 0.0f);
    out[i] = v;
}

// ---------------------------------------------------------------------------
// Launch
// ---------------------------------------------------------------------------
extern "C" void kernel_launch(void* const* d_in, const int* in_sizes, int n_in,
                              void* d_out, int out_size, void* d_ws, size_t ws_size,
                              hipStream_t stream) {
    const float* x    = (const float*)d_in[0];
    const int*   eidx = (const int*)d_in[1];     // [2, E] flattened (int32 under default JAX)
    const float* Wl1  = (const float*)d_in[2];
    const float* Wr1  = (const float*)d_in[3];
    const float* b1   = (const float*)d_in[4];
    const float* Wl2  = (const float*)d_in[5];
    const float* Wr2  = (const float*)d_in[6];
    const float* b2   = (const float*)d_in[7];
    const int* src = eidx;
    const int* dst = eidx + N_EDGES;

    // Workspace layout (floats):
    //  [P : N*64] [R : N*64] [SUMS : N*64] [CNT : N] [H1 : N*64]
    float* P    = (float*)d_ws;
    float* R    = P    + (long long)N_NODES * F_OUT;
    float* SUMS = R    + (long long)N_NODES * F_OUT;
    float* CNT  = SUMS + (long long)N_NODES * F_OUT;
    float* H1   = CNT  + N_NODES;
    float* OUT  = (float*)d_out;

    const int gemm_blocks = N_NODES / 16;        // 6250 (N divisible by 16)
    const long long scat_threads = (long long)N_EDGES * 16;
    const int scat_blocks = (int)((scat_threads + 255) / 256);
    const long long nodef = (long long)N_NODES * F_OUT;
    const int comb_blocks = (int)((nodef + 255) / 256);

    // ---- Layer 1 ----
    {   // zero SUMS and CNT together (contiguous, N*65 floats, /4 exact)
        long long n4 = (long long)N_NODES * (F_OUT + 1) / 4;
        zero_kernel<<<(int)((n4 + 255) / 256), 256, 0, stream>>>(SUMS, n4);
    }
    dual_gemm_kernel<F_IN><<<gemm_blocks, 128, 0, stream>>>(x, Wl1, Wr1, P, R);
    scatter_kernel<true><<<scat_blocks, 256, 0, stream>>>(P, src, dst, SUMS, CNT);
    combine_kernel<true><<<comb_blocks, 256, 0, stream>>>(SUMS, CNT, R, b1, H1);

    // ---- Layer 2 (in-degree counts unchanged; reuse CNT) ----
    {   long long n4 = nodef / 4;
        zero_kernel<<<(int)((n4 + 255) / 256), 256, 0, stream>>>(SUMS, n4);
    }
    dual_gemm_kernel<F_OUT><<<gemm_blocks, 128, 0, stream>>>(H1, Wl2, Wr2, P, R);
    scatter_kernel<false><<<scat_blocks, 256, 0, stream>>>(P, src, dst, SUMS, CNT);
    combine_kernel<false><<<comb_blocks, 256, 0, stream>>>(SUMS, CNT, R, b2, OUT);
}